// ManifoldWorms_20461224198826
// MI455X (gfx1250) — compile-verified
//
#include <hip/hip_runtime.h>
#include <math.h>

typedef __attribute__((ext_vector_type(2))) float v2f;
typedef __attribute__((ext_vector_type(8))) float v8f;

#define C_DIM 1024
#define E_DIM 128
#define I_DIM 8192
#define O_DIM 512
#define U_DIM 64
#define Q_DIM 576   // U + O

// ---------------------------------------------------------------------------
// K1: row-wise L2 normalize (rows x 128). One wave (32 lanes) per row,
// 4 elems/lane, wave32 shfl_xor reduction.
// ---------------------------------------------------------------------------
__global__ void mw_norml2_k(const float* __restrict__ in, float* __restrict__ outp, int rows) {
    int wave = threadIdx.x >> 5;
    int lane = threadIdx.x & 31;
    int row  = blockIdx.x * 8 + wave;
    if (row >= rows) return;
    const float4 v = *(const float4*)(in + row * E_DIM + lane * 4);
    float s = v.x * v.x + v.y * v.y + v.z * v.z + v.w * v.w;
    #pragma unroll
    for (int m = 16; m; m >>= 1) s += __shfl_xor(s, m, 32);
    float inv = 1.0f / sqrtf(s);
    float4 r = make_float4(v.x * inv, v.y * inv, v.z * inv, v.w * inv);
    *(float4*)(outp + row * E_DIM + lane * 4) = r;
}

// ---------------------------------------------------------------------------
// K2: influences = relu(qn @ itn^T).  M=576, N=8192, K=128.
// A = qn (576x128 row-major), B = itn (8192x128 row-major, used transposed).
// Wave tile 32x64: 8 accumulator tiles, B reused across 2 M-subtiles.
// Block = 8 waves = 64(M) x 256(N).  grid(32, 9).
// ---------------------------------------------------------------------------
__global__ void mw_sims_relu_k(const float* __restrict__ A,
                               const float* __restrict__ B,
                               float* __restrict__ Cm) {
    const int K = E_DIM, N = I_DIM;
    int wave = threadIdx.x >> 5, lane = threadIdx.x & 31;
    int mt = blockIdx.y * 64  + (wave >> 2) * 32;
    int nt = blockIdx.x * 256 + (wave & 3) * 64;
    int ml   = lane & 15;       // M (for A) / N (for B) within 16-wide tile
    int lh   = lane >> 4;       // lane half
    int koff = lh * 2;          // K sub-offset per ISA A/B layout

    v8f z = {0.f,0.f,0.f,0.f,0.f,0.f,0.f,0.f};
    v8f acc[2][4] = {{z, z, z, z}, {z, z, z, z}};

    const float* Arow0 = A + (mt + ml) * K + koff;
    const float* Arow1 = Arow0 + 16 * K;
    #pragma unroll 4
    for (int k = 0; k < K; k += 4) {
        v2f a0 = *(const v2f*)(Arow0 + k);
        v2f a1 = *(const v2f*)(Arow1 + k);
        #pragma unroll
        for (int j = 0; j < 4; ++j) {
            v2f b = *(const v2f*)(B + (nt + j * 16 + ml) * K + k + koff);
            acc[0][j] = __builtin_amdgcn_wmma_f32_16x16x4_f32(
                false, a0, false, b, (short)0, acc[0][j], false, false);
            acc[1][j] = __builtin_amdgcn_wmma_f32_16x16x4_f32(
                false, a1, false, b, (short)0, acc[1][j], false, false);
        }
    }
    #pragma unroll
    for (int mi = 0; mi < 2; ++mi)
        #pragma unroll
        for (int j = 0; j < 4; ++j)
            #pragma unroll
            for (int v = 0; v < 8; ++v) {
                int row = mt + mi * 16 + v + lh * 8;
                int col = nt + j * 16 + ml;
                Cm[row * N + col] = fmaxf(acc[mi][j][v], 0.f);
            }
}

// ---------------------------------------------------------------------------
// K3: distributed = influences @ state.  M=576, N=1024, K=8192.
// A = infl (576x8192 row-major), B = state (8192x1024 row-major, lives in L2).
// Wave tile 32x64; B pair-loads reused across 2 M-subtiles -> 8 wmma : 10 loads.
// Rows 0..63 -> unit_in (ws), rows 64..575 -> exit_outputs (d_out).
// grid(4, 9), block 256.
// ---------------------------------------------------------------------------
__global__ void mw_distribute_k(const float* __restrict__ A,
                                const float* __restrict__ B,
                                float* __restrict__ uin,
                                float* __restrict__ exitout) {
    const int K = I_DIM, N = C_DIM;
    int wave = threadIdx.x >> 5, lane = threadIdx.x & 31;
    int mt = blockIdx.y * 64  + (wave >> 2) * 32;
    int nt = blockIdx.x * 256 + (wave & 3) * 64;
    int ml   = lane & 15;
    int lh   = lane >> 4;
    int koff = lh * 2;

    v8f z = {0.f,0.f,0.f,0.f,0.f,0.f,0.f,0.f};
    v8f acc[2][4] = {{z, z, z, z}, {z, z, z, z}};

    const float* Arow0 = A + (mt + ml) * K + koff;
    const float* Arow1 = Arow0 + 16 * K;
    #pragma unroll 2
    for (int k = 0; k < K; k += 4) {
        v2f a0 = *(const v2f*)(Arow0 + k);
        v2f a1 = *(const v2f*)(Arow1 + k);
        #pragma unroll
        for (int j = 0; j < 4; ++j) {
            int col = nt + j * 16 + ml;
            v2f b;
            b.x = B[(k + koff    ) * N + col];
            b.y = B[(k + koff + 1) * N + col];
            acc[0][j] = __builtin_amdgcn_wmma_f32_16x16x4_f32(
                false, a0, false, b, (short)0, acc[0][j], false, false);
            acc[1][j] = __builtin_amdgcn_wmma_f32_16x16x4_f32(
                false, a1, false, b, (short)0, acc[1][j], false, false);
        }
    }
    #pragma unroll
    for (int mi = 0; mi < 2; ++mi)
        #pragma unroll
        for (int j = 0; j < 4; ++j)
            #pragma unroll
            for (int v = 0; v < 8; ++v) {
                int q   = mt + mi * 16 + v + lh * 8;
                int col = nt + j * 16 + ml;
                float val = acc[mi][j][v];
                if (q < U_DIM) uin[q * N + col] = val;
                else           exitout[(q - U_DIM) * N + col] = val;
            }
}

// ---------------------------------------------------------------------------
// K4: garbage[c] = colsum(state)[c] - sum_q distributed[q,c].  grid(4), 256 thr.
// ---------------------------------------------------------------------------
__global__ void mw_garbage_k(const float* __restrict__ state,
                             const float* __restrict__ exitout,
                             const float* __restrict__ uin,
                             float* __restrict__ g) {
    int c = blockIdx.x * 256 + threadIdx.x;   // 0..1023
    float s = 0.f;
    #pragma unroll 8
    for (int i = 0; i < I_DIM; ++i) s += state[i * C_DIM + c];
    float t = 0.f;
    #pragma unroll 4
    for (int q = 0; q < O_DIM; ++q) t += exitout[q * C_DIM + c];
    #pragma unroll 4
    for (int u = 0; u < U_DIM; ++u) t += uin[u * C_DIM + c];
    g[c] = s - t;
}

// ---------------------------------------------------------------------------
// K5: per-unit MLP.  h[u] = unit_in[u] @ W[u] + b[u]; out = unit_in + relu(h).
// Streams 256 MB of W once, coalesced; bandwidth-bound.  grid(4, 64), 256 thr.
// unit_in row staged to LDS via gfx1250 GLOBAL_LOAD_ASYNC_TO_LDS_B128
// (ASYNCcnt-tracked, bypasses VGPRs), then s_wait_asynccnt + barrier.
// ---------------------------------------------------------------------------
__global__ void mw_unitmlp_k(const float* __restrict__ uin,
                             const float* __restrict__ W,
                             const float* __restrict__ bvec,
                             float* __restrict__ outu) {
    __shared__ float s_in[C_DIM];
    int u   = blockIdx.y;
    int tid = threadIdx.x;

    // s_in is the only LDS object in this kernel -> its LDS offset is 0,
    // so the per-lane LDS destination address is simply tid*16 bytes.
    {
        unsigned lds_off = (unsigned)tid * 16u;               // 16B per lane
        const float* gsrc = uin + u * C_DIM + tid * 4;        // 4 floats/lane
        asm volatile("global_load_async_to_lds_b128 %0, %1, off"
                     :: "v"(lds_off), "v"(gsrc)
                     : "memory");
        asm volatile("s_wait_asynccnt 0" ::: "memory");
    }
    __syncthreads();

    int d = blockIdx.x * 256 + tid;
    const float* Wp = W + ((size_t)u << 20) + d;
    float acc = bvec[u * C_DIM + d];
    #pragma unroll 4
    for (int c = 0; c < C_DIM; ++c) acc += s_in[c] * Wp[(size_t)c * C_DIM];
    outu[u * C_DIM + d] = s_in[d] + fmaxf(acc, 0.f);
}

// ---------------------------------------------------------------------------
extern "C" void kernel_launch(void* const* d_in, const int* in_sizes, int n_in,
                              void* d_out, int out_size, void* d_ws, size_t ws_size,
                              hipStream_t stream) {
    const float* state       = (const float*)d_in[0];   // 8192 x 1024
    const float* input_tails = (const float*)d_in[1];   // 8192 x 128
    const float* exit_heads  = (const float*)d_in[2];   // 512 x 128
    const float* unit_heads  = (const float*)d_in[3];   // 64 x 128
    // d_in[4] = unit_tails (unused by reference)
    const float* unit_W      = (const float*)d_in[5];   // 64 x 1024 x 1024
    const float* unit_b      = (const float*)d_in[6];   // 64 x 1024

    float* out      = (float*)d_out;
    float* exit_out = out;                              // 512*1024
    float* garbage  = out + O_DIM * C_DIM;              // 1024
    float* units_o  = garbage + C_DIM;                  // 64*1024

    // workspace layout (floats): ~23.6 MB total
    float* ws   = (float*)d_ws;
    float* itn  = ws;                                   // 8192*128
    float* qn   = itn + I_DIM * E_DIM;                  // 576*128
    float* infl = qn + Q_DIM * E_DIM;                   // 576*8192
    float* uin  = infl + Q_DIM * I_DIM;                 // 64*1024

    // 1. normalize
    mw_norml2_k<<<I_DIM / 8, 256, 0, stream>>>(input_tails, itn, I_DIM);
    mw_norml2_k<<<U_DIM / 8, 256, 0, stream>>>(unit_heads, qn, U_DIM);
    mw_norml2_k<<<O_DIM / 8, 256, 0, stream>>>(exit_heads, qn + U_DIM * E_DIM, O_DIM);

    // 2. influences = relu(qn @ itn^T)   (576 x 8192)
    mw_sims_relu_k<<<dim3(I_DIM / 256, Q_DIM / 64), 256, 0, stream>>>(qn, itn, infl);

    // 3. distributed = influences @ state  -> unit_in + exit_outputs
    mw_distribute_k<<<dim3(C_DIM / 256, Q_DIM / 64), 256, 0, stream>>>(infl, state, uin, exit_out);

    // 4. garbage column sums
    mw_garbage_k<<<C_DIM / 256, 256, 0, stream>>>(state, exit_out, uin, garbage);

    // 5. per-unit MLP
    mw_unitmlp_k<<<dim3(C_DIM / 256, U_DIM), 256, 0, stream>>>(uin, unit_W, unit_b, units_o);
}